// XGate_6992206758256
// MI455X (gfx1250) — compile-verified
//
#include <hip/hip_runtime.h>

// XGate(s=1, index=3, dim=3, N=14), batch=8, f32.
// out[(l,j,r,b)] = x[(l,(j-1)%3,r,b)]  => 81 contiguous slab copies with remap.
//
// Pure permutation: 0 FLOPs, 306 MB traffic -> HBM-bound (~13.1 us at
// 23.3 TB/s; ~6.6 us on graph replay when NT stores keep the 153 MB input
// resident in the 192 MB L2).
//
// CDNA5 async LDS pipe, 8-deep pipelined per wave:
//   8x global_load_async_to_lds_b128 (offset:t*4096, ASYNCcnt=8 in flight)
//   s_wait_asynccnt 0
//   8x global_store_async_from_lds_b128 th:TH_STORE_NT
// The instruction IOFFSET is added to BOTH the global and LDS addresses
// (CDNA5 ISA), so one base VGPR pair + one LDS-offset VGPR
// drive all 8 tiles. Each wave only reads LDS it wrote -> no barrier needed.

typedef float v4f __attribute__((ext_vector_type(4)));

#define DIMQ    3
#define LBLK    27                   // 3^3
#define RBLK    59049                // 3^10
#define BATCH   8
#define RB      (RBLK * BATCH)       // 472392 floats per (l,j) slab
#define CHUNK4  (RB / 4)             // 118098 float4s per slab (16B aligned)
#define NCHUNK  (LBLK * DIMQ)        // 81 slabs
#define TILES   8                    // pipeline depth per wave
#define BTHREADS 256u                // 8 wave32s
#define WIN4    (TILES * BTHREADS)   // 2048 float4s per block window

#define AS_LOAD(OFS)                                                        \
    asm volatile("global_load_async_to_lds_b128 %0, %1, off offset:" #OFS   \
                 :: "v"(lds_off), "v"(src) : "memory")
#define AS_STORE(OFS)                                                       \
    asm volatile("global_store_async_from_lds_b128 %0, %1, off offset:" #OFS\
                 " th:TH_STORE_NT"                                          \
                 :: "v"(dst), "v"(lds_off) : "memory")

// ---------------------------------------------------------------------------
// Primary: 8-deep pipelined copy through the gfx1250 async LDS pipe.
// ---------------------------------------------------------------------------
__global__ void xgate_roll_async(const float* __restrict__ x,
                                 float* __restrict__ out) {
    const unsigned chunk = blockIdx.y;                     // destination slab
    const unsigned j     = chunk % 3u;
    const unsigned src_chunk = chunk + (j == 0u ? 2u : (unsigned)-1);

    // Clamp the whole 2048-float4 window inside the slab; overlapped blocks
    // rewrite identical bytes (same src->dst map), which is benign and keeps
    // EXEC all-ones with uniform strides for the offset:-based pipeline.
    unsigned base = blockIdx.x * WIN4;
    if (base > CHUNK4 - WIN4) base = CHUNK4 - WIN4;
    const unsigned w = base + threadIdx.x;                 // float4 index

    const v4f* src = (const v4f*)x   + (size_t)src_chunk * CHUNK4 + w;
    v4f*       dst = (v4f*)out       + (size_t)chunk     * CHUNK4 + w;

    // Dynamic LDS starts at offset 0: lane slot tid*16, tile t adds t*4096.
    unsigned lds_off = threadIdx.x * 16u;

    // Phase 1: 8 async loads in flight per wave (RT hint: input stays in L2).
    AS_LOAD(0);     AS_LOAD(4096);  AS_LOAD(8192);  AS_LOAD(12288);
    AS_LOAD(16384); AS_LOAD(20480); AS_LOAD(24576); AS_LOAD(28672);
    asm volatile("s_wait_asynccnt 0x0" ::: "memory");

    // Phase 2: 8 async NT stores (streamed output must not evict L2 input).
    AS_STORE(0);     AS_STORE(4096);  AS_STORE(8192);  AS_STORE(12288);
    AS_STORE(16384); AS_STORE(20480); AS_STORE(24576); AS_STORE(28672);
    asm volatile("s_wait_asynccnt 0x0" ::: "memory");
}

// ---------------------------------------------------------------------------
// Fallback/reference path: direct B128 copy, non-temporal store.
// ---------------------------------------------------------------------------
__global__ void xgate_roll_direct(const float* __restrict__ x,
                                  float* __restrict__ out) {
    const unsigned chunk = blockIdx.y;
    const unsigned j     = chunk % 3u;
    const unsigned src_chunk = chunk + (j == 0u ? 2u : (unsigned)-1);

    unsigned w = blockIdx.x * blockDim.x + threadIdx.x;
    if (w >= CHUNK4) return;

    const v4f* src = (const v4f*)x + (size_t)src_chunk * CHUNK4 + w;
    v4f*       dst = (v4f*)out     + (size_t)chunk     * CHUNK4 + w;

    v4f v = *src;
    __builtin_nontemporal_store(v, dst);
}

extern "C" void kernel_launch(void* const* d_in, const int* in_sizes, int n_in,
                              void* d_out, int out_size, void* d_ws, size_t ws_size,
                              hipStream_t stream) {
    const float* x = (const float*)d_in[0];
    float* out     = (float*)d_out;

    dim3 grid((CHUNK4 + WIN4 - 1) / WIN4,   // 58 windows per slab
              NCHUNK, 1);                   // 81 slabs

    // 32 KB dynamic LDS: 256 lanes x 8 tiles x 16 B, LDS offset 0 base.
    xgate_roll_async<<<grid, dim3(BTHREADS, 1, 1), TILES * BTHREADS * 16,
                       stream>>>(x, out);

    (void)in_sizes; (void)n_in; (void)out_size; (void)d_ws; (void)ws_size;
}